// NodeEncoder_32512902430851
// MI455X (gfx1250) — compile-verified
//
#include <hip/hip_runtime.h>

typedef float v2f __attribute__((ext_vector_type(2)));
typedef float v4f __attribute__((ext_vector_type(4)));
typedef float v8f __attribute__((ext_vector_type(8)));

#define N_TYPES       8
#define MAX_DIM       128
#define FEAT          256
#define N_GRAPHS      100000
#define GBLK          (N_GRAPHS / 16)        // 6250 16-graph blocks per type
#define WG_PER_TYPE   64
#define WAVES_PER_WG  8
#define THREADS       (WAVES_PER_WG * 32)
#define WAVES_PER_TYPE (WG_PER_TYPE * WAVES_PER_WG)

// Padded LDS row strides (floats): bank stride 4 -> conflict-free b64 fragment loads
#define WSTRIDE       132
#define XSTRIDE       132
#define W_LDS_FLOATS  (FEAT * WSTRIDE)                 // 33792
#define B_LDS_OFF     W_LDS_FLOATS                     // 256 floats of bias
#define X_LDS_OFF     (B_LDS_OFF + FEAT)
#define X_LDS_PER_WAVE (16 * XSTRIDE)                  // 2112
#define LDS_FLOATS    (X_LDS_OFF + WAVES_PER_WG * X_LDS_PER_WAVE)
#define LDS_BYTES     (LDS_FLOATS * 4)                 // ~203.8 KB < 320 KB/WGP

__global__ __launch_bounds__(THREADS)
void node_encoder_wmma_f32(const float* __restrict__ x,
                           const float* __restrict__ W,
                           const float* __restrict__ b,
                           float* __restrict__ out) {
  extern __shared__ float lds[];

  const int t    = blockIdx.x / WG_PER_TYPE;   // node type
  const int wgi  = blockIdx.x % WG_PER_TYPE;
  const int tid  = threadIdx.x;
  const int lane = tid & 31;
  const int wave = tid >> 5;
  const int rl   = lane & 15;   // row-in-tile (A/M) or col (B,N / D,N)
  const int hi   = lane >> 4;   // lane-half selector (K/M offset)

  const int dims[N_TYPES] = {16, 32, 64, 128, 64, 32, 16, 128};
  const int dim = dims[t];

  // ---- cooperative masked load of W[t] (256x128) into padded LDS ----
  const float* Wt = W + (size_t)t * FEAT * MAX_DIM;
  for (int idx = tid * 4; idx < FEAT * MAX_DIM; idx += THREADS * 4) {
    const int f = idx >> 7;
    const int d = idx & (MAX_DIM - 1);
    v4f wv = *(const v4f*)(Wt + idx);
    if (d >= dim) wv = (v4f){0.f, 0.f, 0.f, 0.f};   // NODE_DIMS column mask
    *(v4f*)&lds[f * WSTRIDE + d] = wv;
  }
  if (tid < FEAT) lds[B_LDS_OFF + tid] = b[t * FEAT + tid];
  __syncthreads();

  float* xs = &lds[X_LDS_OFF + wave * X_LDS_PER_WAVE];
  const int wv = wgi * WAVES_PER_WG + wave;

  for (int gb = wv; gb < GBLK; gb += WAVES_PER_TYPE) {
    const int g0 = gb * 16;

    // ---- stage the 16x128 x tile (coalesced b128 loads -> LDS) ----
#pragma unroll
    for (int r = 0; r < 16; ++r) {
      const float* src = x + ((size_t)(g0 + r) * N_TYPES + t) * MAX_DIM + lane * 4;
      *(v4f*)&xs[r * XSTRIDE + lane * 4] = *(const v4f*)src;
    }

    // ---- all 32 A fragments (16x4 f32 each) for this tile: 64 VGPRs ----
    v2f a[32];
#pragma unroll
    for (int kk = 0; kk < 32; ++kk)
      a[kk] = *(const v2f*)&xs[rl * XSTRIDE + kk * 4 + hi * 2];

    // ---- 16 N-tiles of 16x16; K=128 as 32 chained f32 WMMAs each ----
    for (int nt = 0; nt < 16; ++nt) {
      v8f acc = {0.f, 0.f, 0.f, 0.f, 0.f, 0.f, 0.f, 0.f};
      const float* wrow = &lds[(nt * 16 + rl) * WSTRIDE + hi * 2];
#pragma unroll
      for (int kk = 0; kk < 32; ++kk) {
        const v2f bf = *(const v2f*)(wrow + kk * 4);
        acc = __builtin_amdgcn_wmma_f32_16x16x4_f32(
            /*neg_a=*/false, a[kk], /*neg_b=*/false, bf,
            /*c_mod=*/(short)0, acc, /*reuse_a=*/false, /*reuse_b=*/false);
      }
      const float bias = lds[B_LDS_OFF + nt * 16 + rl];
      // D layout: VGPR j -> M = j + 8*hi, N = rl
      float* dst = out + ((size_t)(g0 + hi * 8) * N_TYPES + t) * FEAT + nt * 16 + rl;
#pragma unroll
      for (int j = 0; j < 8; ++j)
        dst[(size_t)j * N_TYPES * FEAT] = acc[j] + bias;
    }
  }
}

extern "C" void kernel_launch(void* const* d_in, const int* in_sizes, int n_in,
                              void* d_out, int out_size, void* d_ws, size_t ws_size,
                              hipStream_t stream) {
  const float* x = (const float*)d_in[0];   // [800000, 128] f32
  const float* W = (const float*)d_in[1];   // [8, 256, 128] f32
  const float* b = (const float*)d_in[2];   // [8, 256] f32
  float* out = (float*)d_out;               // [800000, 256] f32

  dim3 grid(N_TYPES * WG_PER_TYPE);
  dim3 block(THREADS);
  hipLaunchKernelGGL(node_encoder_wmma_f32, grid, block, LDS_BYTES, stream,
                     x, W, b, out);
}